// GAT_34969623724652
// MI455X (gfx1250) — compile-verified
//
#include <hip/hip_runtime.h>
#include <math.h>

#define GN 50000
#define GE 800000
#define GEI 200000
#define GH 4
#define GD 32
#define GK 128          // feature dim (both layers)
#define GSLOPE 0.2f

typedef __attribute__((ext_vector_type(2))) float v2f;
typedef __attribute__((ext_vector_type(8))) float v8f;

// ---------- helpers ----------
__device__ __forceinline__ unsigned f2key(float f) {
    int i = __float_as_int(f);
    return (unsigned)(i ^ ((i >> 31) | 0x80000000));
}
__device__ __forceinline__ float key2f(unsigned k) {
    int i = (k & 0x80000000u) ? (int)(k ^ 0x80000000u) : ~(int)k;
    return __int_as_float(i);
}
// Guaranteed-hardware f32 atomic add (non-returning -> STOREcnt), device scope.
__device__ __forceinline__ void atomAddF(float* p, float v) {
    asm volatile("global_atomic_add_f32 %0, %1, off scope:SCOPE_DEV"
                 :: "v"(p), "v"(v) : "memory");
}

// ---------- GEMM: C[N x 128] = A[N x 128] * W[128 x 128], f32 WMMA ----------
// One block = 256 threads = 8 waves. Block b computes rows [16b, 16b+16).
// Wave w computes the 16x16 tile at columns [16w, 16w+16). K swept in steps of 4.
__global__ __launch_bounds__(256) void gat_gemm_wmma(
    const float* __restrict__ A, const float* __restrict__ W, float* __restrict__ C)
{
    __shared__ float lA[16 * GK];   // 8 KB A tile
    const int tid   = threadIdx.x;
    const int tileM = blockIdx.x;

    // cooperative load of the 16x128 A tile (512 float4 / 256 threads)
    const float4* A4 = (const float4*)(A + (size_t)tileM * 16 * GK);
    float4* lA4 = (float4*)lA;
    for (int i = tid; i < 16 * GK / 4; i += 256) lA4[i] = A4[i];
    __syncthreads();

    const int wave  = tid >> 5;          // 0..7 -> column tile
    const int lane  = tid & 31;
    const int row   = lane & 15;         // M within tile (A frag)
    const int khalf = (lane >> 4) << 1;  // 0 or 2
    const int col   = wave * 16 + (lane & 15);  // global column (B/C frag)

    v8f c = {0.f, 0.f, 0.f, 0.f, 0.f, 0.f, 0.f, 0.f};
    #pragma unroll 4
    for (int k0 = 0; k0 < GK; k0 += 4) {
        v2f a, b;
        a.x = lA[row * GK + k0 + khalf];
        a.y = lA[row * GK + k0 + khalf + 1];
        b.x = W[(k0 + khalf)     * GK + col];
        b.y = W[(k0 + khalf + 1) * GK + col];
        c = __builtin_amdgcn_wmma_f32_16x16x4_f32(
                /*neg_a=*/false, a, /*neg_b=*/false, b,
                /*c_mod=*/(short)0, c, /*reuse_a=*/false, /*reuse_b=*/false);
    }

    const int rowBase = tileM * 16 + (lane >> 4) * 8;   // M = vgpr + 8*(lane>=16)
    #pragma unroll
    for (int v = 0; v < 8; ++v)
        C[(size_t)(rowBase + v) * GK + col] = c[v];
}

// ---------- el/er: per (node, head) dot of ft with al/ar ----------
__global__ __launch_bounds__(256) void gat_elr(
    const float* __restrict__ ft, const float* __restrict__ al,
    const float* __restrict__ ar, float* __restrict__ el, float* __restrict__ er)
{
    int i = blockIdx.x * 256 + threadIdx.x;        // over N*H
    if (i >= GN * GH) return;
    int n = i >> 2, h = i & 3;
    const float* f  = ft + (size_t)n * GK + h * GD;
    const float* pa = al + h * GD;
    const float* pb = ar + h * GD;
    float sl = 0.f, sr = 0.f;
    #pragma unroll
    for (int j = 0; j < GD; ++j) { float v = f[j]; sl += v * pa[j]; sr += v * pb[j]; }
    el[i] = sl; er[i] = sr;
}

// ---------- init accumulators ----------
__global__ __launch_bounds__(256) void gat_init(
    float* __restrict__ msg, unsigned* __restrict__ mkey, float* __restrict__ denom)
{
    int i = blockIdx.x * 256 + threadIdx.x;        // over N*128
    if (i < GN * GK) msg[i] = 0.f;
    if (i < GN * GH) { mkey[i] = 0u; denom[i] = 0.f; }
}

// ---------- edge pass 1: segment max of leaky_relu(el[src]+er[dst]) ----------
__global__ __launch_bounds__(256) void gat_edge_max(
    const int* __restrict__ src, const int* __restrict__ dst,
    const float* __restrict__ el, const float* __restrict__ er,
    unsigned* __restrict__ mkey)
{
    int i = blockIdx.x * 256 + threadIdx.x;        // over E*H
    if (i >= GE * GH) return;
    int e = i >> 2, h = i & 3;
    int s = src[e], d = dst[e];
    float x = el[s * GH + h] + er[d * GH + h];
    x = x > 0.f ? x : GSLOPE * x;
    atomicMax(&mkey[d * GH + h], f2key(x));
}

// ---------- edge pass 2: a = exp(s - m[dst]) once per (edge,head); denom += a ----------
__global__ __launch_bounds__(256) void gat_edge_coef(
    const int* __restrict__ src, const int* __restrict__ dst,
    const float* __restrict__ el, const float* __restrict__ er,
    const unsigned* __restrict__ mkey, float* __restrict__ acoef,
    float* __restrict__ denom)
{
    int i = blockIdx.x * 256 + threadIdx.x;        // over E*H
    if (i >= GE * GH) return;
    int e = i >> 2, h = i & 3;
    int s = src[e], d = dst[e];
    float x = el[s * GH + h] + er[d * GH + h];
    x = x > 0.f ? x : GSLOPE * x;
    float a = expf(x - key2f(mkey[d * GH + h]));
    acoef[i] = a;
    atomAddF(&denom[d * GH + h], a);
}

// ---------- edge pass 3: msg[dst] += a * ft[src]  (pure scatter stream) ----------
__global__ __launch_bounds__(256) void gat_scatter(
    const int* __restrict__ src, const int* __restrict__ dst,
    const float* __restrict__ acoef, const float* __restrict__ ft,
    float* __restrict__ msg)
{
    int i = blockIdx.x * 256 + threadIdx.x;        // over E*128
    if (i >= GE * GK) return;
    int e = i >> 7, f = i & 127, h = f >> 5;
    int s = src[e], d = dst[e];
    float a = acoef[e * GH + h];
    atomAddF(&msg[(size_t)d * GK + f], a * ft[(size_t)s * GK + f]);
}

// ---------- normalize + ELU ----------
__global__ __launch_bounds__(256) void gat_norm(
    const float* __restrict__ msg, const float* __restrict__ denom,
    float* __restrict__ hout)
{
    int i = blockIdx.x * 256 + threadIdx.x;        // over N*128
    if (i >= GN * GK) return;
    int n = i >> 7, h = (i >> 5) & 3;
    float r = msg[i] / fmaxf(denom[n * GH + h], 1e-9f);
    hout[i] = r > 0.f ? r : expm1f(r);
}

// ---------- final gather: edges[E_IND, H*2D] ----------
__global__ __launch_bounds__(256) void gat_gather(
    const float* __restrict__ h, const int* __restrict__ i0,
    const int* __restrict__ i1, float* __restrict__ out)
{
    int i = blockIdx.x * 256 + threadIdx.x;        // over E_IND*256
    if (i >= GEI * 2 * GK) return;
    int row = i >> 8, f = i & 255;
    int head = f >> 6, half = (f >> 5) & 1, d = f & 31;
    int node = half ? i1[row] : i0[row];
    out[i] = h[(size_t)node * GK + head * GD + d];
}

// ---------- host-side layer driver ----------
static void run_gat_layer(const float* hin, const float* W, const float* al, const float* ar,
                          const int* src, const int* dst,
                          float* ft, float* el, float* er, unsigned* mkey,
                          float* denom, float* msg, float* acoef, float* hout,
                          hipStream_t stream)
{
    gat_gemm_wmma<<<GN / 16, 256, 0, stream>>>(hin, W, ft);
    gat_elr<<<(GN * GH + 255) / 256, 256, 0, stream>>>(ft, al, ar, el, er);
    gat_init<<<(GN * GK + 255) / 256, 256, 0, stream>>>(msg, mkey, denom);
    gat_edge_max<<<(GE * GH + 255) / 256, 256, 0, stream>>>(src, dst, el, er, mkey);
    gat_edge_coef<<<(GE * GH + 255) / 256, 256, 0, stream>>>(src, dst, el, er, mkey, acoef, denom);
    gat_scatter<<<(GE * GK + 255) / 256, 256, 0, stream>>>(src, dst, acoef, ft, msg);
    gat_norm<<<(GN * GK + 255) / 256, 256, 0, stream>>>(msg, denom, hout);
}

extern "C" void kernel_launch(void* const* d_in, const int* in_sizes, int n_in,
                              void* d_out, int out_size, void* d_ws, size_t ws_size,
                              hipStream_t stream)
{
    const float* x      = (const float*)d_in[0];
    const int*   src    = (const int*)  d_in[1];
    const int*   dst    = (const int*)  d_in[2];
    const int*   indics = (const int*)  d_in[3];
    const float* W0     = (const float*)d_in[4];
    const float* al0    = (const float*)d_in[5];
    const float* ar0    = (const float*)d_in[6];
    const float* W1     = (const float*)d_in[7];
    const float* al1    = (const float*)d_in[8];
    const float* ar1    = (const float*)d_in[9];

    // workspace carve-up (floats): ft | h0 | msg | el | er | denom | mkey | acoef
    float* ws    = (float*)d_ws;
    float* ft    = ws;
    float* h0    = ft  + (size_t)GN * GK;
    float* msg   = h0  + (size_t)GN * GK;
    float* el    = msg + (size_t)GN * GK;
    float* er    = el  + (size_t)GN * GH;
    float* denom = er  + (size_t)GN * GH;
    unsigned* mkey = (unsigned*)(denom + (size_t)GN * GH);
    float* acoef = (float*)(mkey + (size_t)GN * GH);     // E*H floats

    float* hfinal = (float*)d_out;                       // [N, H*D]
    float* edges  = hfinal + (size_t)GN * GK;            // [E_IND, H*2D]

    // layer 0: x -> h0 (workspace)
    run_gat_layer(x, W0, al0, ar0, src, dst, ft, el, er, mkey, denom, msg, acoef, h0, stream);
    // layer 1: h0 -> hfinal (d_out)
    run_gat_layer(h0, W1, al1, ar1, src, dst, ft, el, er, mkey, denom, msg, acoef, hfinal, stream);
    // edge feature gather
    gat_gather<<<(GEI * 2 * GK + 255) / 256, 256, 0, stream>>>(
        hfinal, indics, indics + GEI, edges);
}